// LR_23029614641373
// MI455X (gfx1250) — compile-verified
//
#include <hip/hip_runtime.h>
#include <hip/hip_bf16.h>
#include <cstdint>

// Logistic regression over one-hot(user) ++ one-hot(movie):
//   logit = w[u] + w[N_USERS + m] + b ;  p = sigmoid(logit) ; out = [1-p, p]
// Memory-bound: 16 B HBM per sample. W (9923 f32 = 39.7 KB) staged in LDS via
// CDNA5 async global->LDS instructions; gathers are LDS-local.

#define N_USERS   6040
#define N_MOVIES  3883
#define W_LEN     (N_USERS + N_MOVIES)   // 9923 floats
#define W_B128    (W_LEN / 4)            // 2480 full 16-byte chunks (9920 floats)
#define W_TAIL    (W_LEN - 4 * W_B128)   // 3 tail floats

__global__ __launch_bounds__(256) void lr_onehot_kernel(
    const int* __restrict__ x,      // [B,2] interleaved (user, movie)
    const float* __restrict__ W,    // [9923]
    const float* __restrict__ bptr, // [1]
    float* __restrict__ out,        // [B,2] -> (1-p, p)
    int nQuads)                     // number of int4 chunks of x (2 samples each)
{
    __shared__ float sW[W_LEN + 1]; // +1 pad

    // ---- Stage W into LDS with CDNA5 async global->LDS copies (ASYNCcnt) ----
    const uint32_t lds_base = (uint32_t)(uintptr_t)(&sW[0]); // low 32 bits = LDS offset
    for (int c = threadIdx.x; c < W_B128; c += blockDim.x) {
        uint32_t goff = (uint32_t)c * 16u;
        uint32_t loff = lds_base + goff;
        asm volatile("global_load_async_to_lds_b128 %0, %1, %2"
                     :: "v"(loff), "v"(goff), "s"(W) : "memory");
    }
    if (threadIdx.x < W_TAIL) {
        uint32_t goff = (uint32_t)(4 * W_B128 + threadIdx.x) * 4u;
        uint32_t loff = lds_base + goff;
        asm volatile("global_load_async_to_lds_b32 %0, %1, %2"
                     :: "v"(loff), "v"(goff), "s"(W) : "memory");
    }
    asm volatile("s_wait_asynccnt 0x0" ::: "memory");
    __syncthreads();

    const float bias = bptr[0];

    const int4*  x4 = (const int4*)x;   // {u0, m0, u1, m1}
    float4*      o4 = (float4*)out;     // {1-p0, p0, 1-p1, p1}

    const int stride = gridDim.x * blockDim.x;
    for (int i = blockIdx.x * blockDim.x + threadIdx.x; i < nQuads; i += stride) {
        __builtin_prefetch(&x4[i + stride], 0, 0);   // global_prefetch_b8

        int4 xm = x4[i];                             // 16 B: two samples

        // LDS gathers (bank-random over 64 banks; vmem path stays free)
        float z0 = sW[xm.x] + sW[N_USERS + xm.y] + bias;
        float z1 = sW[xm.z] + sW[N_USERS + xm.w] + bias;

        // sigmoid: t = e^-z ; p = 1/(1+t) ; 1-p = t/(1+t) = t*p
        float t0 = __expf(-z0);
        float t1 = __expf(-z1);
        float p0 = __builtin_amdgcn_rcpf(1.0f + t0);
        float p1 = __builtin_amdgcn_rcpf(1.0f + t1);

        float4 r;
        r.x = t0 * p0;  r.y = p0;
        r.z = t1 * p1;  r.w = p1;
        o4[i] = r;                                   // 16 B store
    }
}

extern "C" void kernel_launch(void* const* d_in, const int* in_sizes, int n_in,
                              void* d_out, int out_size, void* d_ws, size_t ws_size,
                              hipStream_t stream) {
    (void)n_in; (void)out_size; (void)d_ws; (void)ws_size;
    const int*   x = (const int*)d_in[0];    // [B,2] int32
    const float* W = (const float*)d_in[1];  // [1, 9923] f32
    const float* b = (const float*)d_in[2];  // [1] f32
    float* out = (float*)d_out;              // [B,2] f32

    const int B      = in_sizes[0] / 2;      // samples
    const int nQuads = B / 2;                // int4 chunks (B is even: 4194304)

    const int threads = 256;                 // 8 wave32s per block
    const int blocks  = 2048;                // grid-stride; ~16K waves of MLP
    lr_onehot_kernel<<<blocks, threads, 0, stream>>>(x, W, b, out, nQuads);
}